// HausdorffDTLoss_86328842649638
// MI455X (gfx1250) — compile-verified
//
#include <hip/hip_runtime.h>
#include <math.h>

// Problem constants (match reference setup_inputs)
#define Bn 8
#define Cn 4
#define Hn 256
#define Wn 256
constexpr float BIGF = (float)(Hn + Wn);   // 512
constexpr int HW   = Hn * Wn;              // 65536
constexpr int BCHW = Bn * Cn * Hn * Wn;    // 2097152

// Workspace layout (bytes)
constexpr size_t OFF_MASK  = 0;                                    // Bn*HW bytes
constexpr size_t OFF_G2    = (size_t)Bn * HW;                      // 4*BCHW floats
constexpr size_t OFF_FLAGS = OFF_G2 + (size_t)4 * BCHW * 4;        // 128 ints
constexpr size_t OFF_PART  = OFF_FLAGS + 512;                      // Cn*Bn*Hn floats

typedef __attribute__((ext_vector_type(2))) float v2f;
typedef __attribute__((ext_vector_type(8))) float v8f;

// ---------------------------------------------------------------------------
// Kernel A: softmax -> packed binary masks (bits 0..3 = p_bin per class,
// bits 4..7 = onehot per class); also zero the "mask any" flags.
// ---------------------------------------------------------------------------
__global__ __launch_bounds__(256) void hd_mask_kernel(
    const float* __restrict__ logits, const int* __restrict__ target,
    unsigned char* __restrict__ mask, int* __restrict__ flags) {
  int gid = blockIdx.x * 256 + threadIdx.x;
  if (gid < 4 * Bn * Cn) flags[gid] = 0;
  if (gid >= Bn * HW) return;
  int b   = gid / HW;
  int rem = gid - b * HW;
  float l[Cn];
#pragma unroll
  for (int cc = 0; cc < Cn; ++cc) l[cc] = logits[(size_t)(b * Cn + cc) * HW + rem];
  float mx = l[0];
#pragma unroll
  for (int cc = 1; cc < Cn; ++cc) mx = fmaxf(mx, l[cc]);
  float e[Cn], sum = 0.0f;
#pragma unroll
  for (int cc = 0; cc < Cn; ++cc) { e[cc] = expf(l[cc] - mx); sum += e[cc]; }
  float inv = 1.0f / sum;
  int t = target[gid];
  unsigned int bits = 0;
#pragma unroll
  for (int cc = 0; cc < Cn; ++cc) {
    if (e[cc] * inv > 0.5f) bits |= (1u << cc);
    if (t == cc)            bits |= (1u << (4 + cc));
  }
  mask[gid] = (unsigned char)bits;
}

// ---------------------------------------------------------------------------
// Kernel B: vertical 1D distance (two scans) for 4 mask families at once.
// One thread per (b,c,w) column. Writes g^2; sets per-mask any-flags.
// Families: 0 = p_bin, 1 = ~p_bin, 2 = g_bin, 3 = ~g_bin.
// ---------------------------------------------------------------------------
__global__ __launch_bounds__(256) void hd_vert_kernel(
    const unsigned char* __restrict__ mask, float* __restrict__ g2,
    int* __restrict__ flags) {
  int bc = blockIdx.x;              // 0..Bn*Cn-1
  int b  = bc / Cn;
  int c  = bc - b * Cn;
  int w  = threadIdx.x;
  const unsigned char* mcol = mask + (size_t)b * HW + w;
  size_t base = (size_t)(bc * Hn) * Wn + w;

  float last0 = -BIGF, last1 = -BIGF, last2 = -BIGF, last3 = -BIGF;
  int any0 = 0, any1 = 0, any2 = 0, any3 = 0;

  // forward pass: store (h - lastTrueAbove) as temp
  for (int h = 0; h < Hn; ++h) {
    unsigned int mb = mcol[(size_t)h * Wn];
    float fh = (float)h;
    bool m0 = (mb >> c) & 1u;
    bool m2 = (mb >> (4 + c)) & 1u;
    if (m0)  { last0 = fh; any0 = 1; } else { last1 = fh; any1 = 1; }
    if (m2)  { last2 = fh; any2 = 1; } else { last3 = fh; any3 = 1; }
    size_t idx = base + (size_t)h * Wn;
    g2[idx]                      = fh - last0;
    g2[idx + (size_t)BCHW]       = fh - last1;
    g2[idx + 2 * (size_t)BCHW]   = fh - last2;
    g2[idx + 3 * (size_t)BCHW]   = fh - last3;
  }
  // backward pass: combine with nextTrueBelow, clamp to BIG, square
  float nxt0 = 2.0f * BIGF, nxt1 = 2.0f * BIGF, nxt2 = 2.0f * BIGF, nxt3 = 2.0f * BIGF;
  for (int h = Hn - 1; h >= 0; --h) {
    unsigned int mb = mcol[(size_t)h * Wn];
    float fh = (float)h;
    bool m0 = (mb >> c) & 1u;
    bool m2 = (mb >> (4 + c)) & 1u;
    if (m0) nxt0 = fh; else nxt1 = fh;
    if (m2) nxt2 = fh; else nxt3 = fh;
    size_t idx = base + (size_t)h * Wn;
    float g;
    g = fminf(fminf(g2[idx],                    nxt0 - fh), BIGF); g2[idx]                    = g * g;
    g = fminf(fminf(g2[idx + (size_t)BCHW],     nxt1 - fh), BIGF); g2[idx + (size_t)BCHW]     = g * g;
    g = fminf(fminf(g2[idx + 2*(size_t)BCHW],   nxt2 - fh), BIGF); g2[idx + 2*(size_t)BCHW]   = g * g;
    g = fminf(fminf(g2[idx + 3*(size_t)BCHW],   nxt3 - fh), BIGF); g2[idx + 3*(size_t)BCHW]   = g * g;
  }
  if (any0) atomicOr(&flags[0 * Bn * Cn + bc], 1);
  if (any1) atomicOr(&flags[1 * Bn * Cn + bc], 1);
  if (any2) atomicOr(&flags[2 * Bn * Cn + bc], 1);
  if (any3) atomicOr(&flags[3 * Bn * Cn + bc], 1);
}

// ---------------------------------------------------------------------------
// Kernel C: per-row lower envelope for all 4 families + fused loss terms.
// One block per (b,c,h). d2[i] = min_j ((i-j)^2 + g2[j]).
// Hot loop uses float4 LDS reads (ds_load_b128, broadcast address across the
// wave) so the min-plus chain is VALU-issue bound, not DS-issue bound.
// ---------------------------------------------------------------------------
__global__ __launch_bounds__(256) void hd_row_kernel(
    const float* __restrict__ g2, const float* __restrict__ logits,
    const int* __restrict__ target, const int* __restrict__ flags,
    float* __restrict__ partials) {
  __shared__ float4 s0[Wn / 4], s1[Wn / 4], s2[Wn / 4], s3[Wn / 4];
  __shared__ float red[256];
  int bch = blockIdx.x;
  int h  = bch % Hn;
  int bc = bch / Hn;
  int c  = bc % Cn;
  int b  = bc / Cn;
  int i  = threadIdx.x;
  size_t rowbase = (size_t)(bc * Hn + h) * Wn;
  ((float*)s0)[i] = g2[rowbase + i];
  ((float*)s1)[i] = g2[rowbase + i + (size_t)BCHW];
  ((float*)s2)[i] = g2[rowbase + i + 2 * (size_t)BCHW];
  ((float*)s3)[i] = g2[rowbase + i + 3 * (size_t)BCHW];
  __syncthreads();

  float d20 = 3.0e38f, d21 = 3.0e38f, d22 = 3.0e38f, d23 = 3.0e38f;
  float fi = (float)i;
#pragma unroll 2
  for (int jv = 0; jv < Wn / 4; ++jv) {
    float4 a0 = s0[jv];
    float4 a1 = s1[jv];
    float4 a2 = s2[jv];
    float4 a3 = s3[jv];
    float t0 = fi - (float)(4 * jv);
    float t1 = t0 - 1.0f;
    float t2 = t0 - 2.0f;
    float t3 = t0 - 3.0f;
    float q0 = t0 * t0, q1 = t1 * t1, q2 = t2 * t2, q3 = t3 * t3;
    d20 = fminf(d20, q0 + a0.x); d20 = fminf(d20, q1 + a0.y);
    d20 = fminf(d20, q2 + a0.z); d20 = fminf(d20, q3 + a0.w);
    d21 = fminf(d21, q0 + a1.x); d21 = fminf(d21, q1 + a1.y);
    d21 = fminf(d21, q2 + a1.z); d21 = fminf(d21, q3 + a1.w);
    d22 = fminf(d22, q0 + a2.x); d22 = fminf(d22, q1 + a2.y);
    d22 = fminf(d22, q2 + a2.z); d22 = fminf(d22, q3 + a2.w);
    d23 = fminf(d23, q0 + a3.x); d23 = fminf(d23, q1 + a3.y);
    d23 = fminf(d23, q2 + a3.z); d23 = fminf(d23, q3 + a3.w);
  }
  // empty-mask fallback: edt == 0 everywhere
  float d0  = flags[bc]             ? sqrtf(d20) : 0.0f;
  float d1  = flags[Bn*Cn + bc]     ? sqrtf(d21) : 0.0f;
  float d2v = flags[2*Bn*Cn + bc]   ? sqrtf(d22) : 0.0f;
  float d3  = flags[3*Bn*Cn + bc]   ? sqrtf(d23) : 0.0f;
  float dtp = d0 + d1, dtg = d2v + d3;
  float wgt = dtp * dtp + dtg * dtg;     // alpha = 2

  // recompute softmax prob for class c at this pixel
  size_t roff = (size_t)h * Wn + i;
  float l0 = logits[(size_t)(b * Cn + 0) * HW + roff];
  float l1 = logits[(size_t)(b * Cn + 1) * HW + roff];
  float l2 = logits[(size_t)(b * Cn + 2) * HW + roff];
  float l3 = logits[(size_t)(b * Cn + 3) * HW + roff];
  float mx = fmaxf(fmaxf(l0, l1), fmaxf(l2, l3));
  float e0 = expf(l0 - mx), e1 = expf(l1 - mx), e2 = expf(l2 - mx), e3 = expf(l3 - mx);
  float inv = 1.0f / (e0 + e1 + e2 + e3);
  float lc = (c == 0) ? l0 : (c == 1) ? l1 : (c == 2) ? l2 : l3;
  float p  = expf(lc - mx) * inv;
  int tg = target[(size_t)b * HW + roff];
  float oh = (tg == c) ? 1.0f : 0.0f;
  float diff = p - oh;
  float val = diff * diff * wgt;

  red[i] = val;
  __syncthreads();
#pragma unroll
  for (int s = 128; s > 0; s >>= 1) {
    if (i < s) red[i] += red[i + s];
    __syncthreads();
  }
  if (i == 0) partials[(size_t)c * (Bn * Hn) + b * Hn + h] = red[0];
}

// ---------------------------------------------------------------------------
// Kernel D: per-class reduction of 2048 partials via V_WMMA_F32_16X16X4_F32
// with an all-ones B matrix (exact f32 sums on the matrix unit), then the
// final scalar. 128 threads = 4 waves, one wave per class; uniform control
// flow inside each wave keeps EXEC all-ones around the WMMA.
// ---------------------------------------------------------------------------
__global__ __launch_bounds__(128) void hd_final_kernel(
    const float* __restrict__ partials, float* __restrict__ out) {
  __shared__ float cls[Cn];
  int lane = threadIdx.x & 31;
  int c    = threadIdx.x >> 5;       // wave id == class
  const int N = Bn * Hn;             // 2048 partials per class
  const float* p = partials + (size_t)c * N;

  // A is 16x4 f32: lane m<16 holds A[m][0..1], lane m+16 holds A[m][2..3]
  int m  = lane & 15;
  int kb = (lane >> 4) * 2;
  v2f ones; ones[0] = 1.0f; ones[1] = 1.0f;
  v8f acc = {};
  for (int base = 0; base < N; base += 64) {     // 32 uniform iterations
    v2f a;
    a[0] = p[base + m * 4 + kb];
    a[1] = p[base + m * 4 + kb + 1];
    // D[m,n] = sum_k A[m,k]*1 + C[m,n]  (exact f32 accumulation)
    acc = __builtin_amdgcn_wmma_f32_16x16x4_f32(
        /*neg_a=*/false, a, /*neg_b=*/false, ones,
        /*c_mod=*/(short)0, acc, /*reuse_a=*/false, /*reuse_b=*/false);
  }
  float s = 0.0f;
#pragma unroll
  for (int k = 0; k < 8; ++k) s += acc[k];
  // column n=0 lives in lanes 0 (M=0..7) and 16 (M=8..15)
  float sum_lo = __shfl(s, 0, 32);
  float sum_hi = __shfl(s, 16, 32);
  if (lane == 0) cls[c] = sum_lo + sum_hi;
  __syncthreads();
  if (threadIdx.x == 0) {
    float denom = (float)(Bn * Hn * Wn) + 1e-6f;
    float tot = 0.0f;
#pragma unroll
    for (int cc = 0; cc < Cn; ++cc) tot += cls[cc] / denom;
    out[0] = tot / (float)Cn;
  }
}

// ---------------------------------------------------------------------------
extern "C" void kernel_launch(void* const* d_in, const int* in_sizes, int n_in,
                              void* d_out, int out_size, void* d_ws, size_t ws_size,
                              hipStream_t stream) {
  const float* logits = (const float*)d_in[0];   // [B,C,H,W] f32
  const int*   target = (const int*)d_in[1];     // [B,H,W] i32
  float* out = (float*)d_out;                    // scalar f32

  unsigned char* ws  = (unsigned char*)d_ws;
  unsigned char* mask = ws + OFF_MASK;
  float* g2       = (float*)(ws + OFF_G2);
  int*   flags    = (int*)(ws + OFF_FLAGS);
  float* partials = (float*)(ws + OFF_PART);

  (void)in_sizes; (void)n_in; (void)out_size; (void)ws_size;

  hd_mask_kernel<<<(Bn * HW) / 256, 256, 0, stream>>>(logits, target, mask, flags);
  hd_vert_kernel<<<Bn * Cn, 256, 0, stream>>>(mask, g2, flags);
  hd_row_kernel<<<Bn * Cn * Hn, 256, 0, stream>>>(g2, logits, target, flags, partials);
  hd_final_kernel<<<1, 128, 0, stream>>>(partials, out);
}